// MultiHeadAdditiveAttention_75909251989607
// MI455X (gfx1250) — compile-verified
//
#include <hip/hip_runtime.h>
#include <math.h>

// Problem constants (match reference)
#define BB   2
#define LL   384
#define DD   512
#define HH   8
#define DKN  64
#define RR   (BB * LL)   // 768 rows

typedef __attribute__((ext_vector_type(2))) float v2f;
typedef __attribute__((ext_vector_type(4))) float v4f;
typedef __attribute__((ext_vector_type(8))) float v8f;

typedef __attribute__((address_space(1))) int as1_int;   // global
typedef __attribute__((address_space(3))) int as3_int;   // LDS

// V_WMMA_F32_16X16X4_F32: D = A(16x4,f32) * B(4x16,f32) + C(16x16,f32)
#define WMMA_F32_4(a, b, c) \
  __builtin_amdgcn_wmma_f32_16x16x4_f32(false, (a), false, (b), (short)0, (c), false, false)

__device__ __forceinline__ float fast_tanh(float x) {
#if __has_builtin(__builtin_amdgcn_tanhf)
  return __builtin_amdgcn_tanhf(x);   // v_tanh_f32 (gfx1250 TRANS pipe)
#else
  return tanhf(x);
#endif
}

// ---------------------------------------------------------------------------
// gfx1250 async global->LDS copy (ASYNCcnt path), with safe fallback.
// ---------------------------------------------------------------------------
#if __has_builtin(__builtin_amdgcn_global_load_async_to_lds_b32)
#define HAVE_ASYNC_LDS 1
#else
#define HAVE_ASYNC_LDS 0
#endif

__device__ __forceinline__ void stage_b32_to_lds(const float* g, float* l) {
#if HAVE_ASYNC_LDS
  __builtin_amdgcn_global_load_async_to_lds_b32(
      (as1_int*)g, (as3_int*)l, /*offset=*/0, /*cpol=*/0);
#else
  *l = *g;
#endif
}

__device__ __forceinline__ void wait_async_lds() {
#if HAVE_ASYNC_LDS
#if __has_builtin(__builtin_amdgcn_s_wait_asynccnt)
  __builtin_amdgcn_s_wait_asynccnt(0);
#else
  asm volatile("s_wait_asynccnt 0x0" ::: "memory");
#endif
#endif
}

// ---------------------------------------------------------------------------
// C[R x 512] = A[R x 512] @ W^T (W is [512 x 512], row-major) + bias[512]
// Block = 128 threads = 4 waves. blockIdx.x -> 16-row M tile,
// blockIdx.y*64 + wave*16 -> N tile. K loop of 512 in steps of 4 (WMMA f32).
// ---------------------------------------------------------------------------
__global__ __launch_bounds__(128) void gemm_bias_512(
    const float* __restrict__ A, const float* __restrict__ W,
    const float* __restrict__ bias, float* __restrict__ C)
{
  const int lane  = threadIdx.x & 31;
  const int wave  = threadIdx.x >> 5;
  const int mbase = blockIdx.x * 16;
  const int nbase = blockIdx.y * 64 + wave * 16;
  const int mrow  = mbase + (lane & 15);
  const int ncol  = nbase + (lane & 15);
  const int khalf = (lane >> 4) * 2;       // lanes 0-15 -> K 0,1 ; lanes 16-31 -> K 2,3

  const float* arow = A + mrow * DD;
  const float* wrow = W + ncol * DD;       // row n of W == column n of W^T

  v8f acc = {};
  for (int kk = 0; kk < DD; kk += 4) {
    v2f a = *(const v2f*)(arow + kk + khalf);
    v2f b = *(const v2f*)(wrow + kk + khalf);
    acc = WMMA_F32_4(a, b, acc);
  }

  const float bv = bias[ncol];
#pragma unroll
  for (int r = 0; r < 8; ++r) {
    const int row = (lane >> 4) * 8 + r;   // C/D layout: VGPR r -> M = r (+8 for hi half)
    C[(mbase + row) * DD + ncol] = acc[r] + bv;
  }
}

// ---------------------------------------------------------------------------
// Per-head 64x64 projection: Y[r, h*64+i] = sum_j X[r, h*64+j]*Wh[i*64+j] + bh[i]
// blockIdx.x -> 16-row M tile, blockIdx.y -> head; 4 waves cover the 4 N-tiles.
// ---------------------------------------------------------------------------
__global__ __launch_bounds__(128) void head_gemm_64(
    const float* __restrict__ X, const float* __restrict__ Wh,
    const float* __restrict__ bh, float* __restrict__ Y)
{
  const int lane  = threadIdx.x & 31;
  const int wave  = threadIdx.x >> 5;      // N-tile within head (0..3)
  const int mbase = blockIdx.x * 16;
  const int h     = blockIdx.y;
  const int nh    = wave * 16 + (lane & 15);  // column within head (0..63)
  const int khalf = (lane >> 4) * 2;

  const float* xrow = X + (mbase + (lane & 15)) * DD + h * DKN;
  const float* wrow = Wh + nh * DKN;

  v8f acc = {};
  for (int kk = 0; kk < DKN; kk += 4) {
    v2f a = *(const v2f*)(xrow + kk + khalf);
    v2f b = *(const v2f*)(wrow + kk + khalf);
    acc = WMMA_F32_4(a, b, acc);
  }

  const float bv = bh[nh];
#pragma unroll
  for (int r = 0; r < 8; ++r) {
    const int row = (lane >> 4) * 8 + r;
    Y[(mbase + row) * DD + h * DKN + nh] = acc[r] + bv;
  }
}

// ---------------------------------------------------------------------------
// Fused additive attention for one (b, h, 16-query tile):
//   scores[q,k] = sum_d va[d] * tanh(Qa[q,d] + Ka[k,d])   (TRANS-pipe bound)
//   P = softmax_k(scores);  ctx[q,:] = P @ V              (WMMA f32 16x16x4)
// Buffers are flat [b, l, h, dk] (= [B*L, 512]), so no transposes needed.
// Qa tile + va are staged into LDS with gfx1250 async global->LDS copies.
// ---------------------------------------------------------------------------
__global__ __launch_bounds__(128) void additive_attn(
    const float* __restrict__ Qa, const float* __restrict__ Ka,
    const float* __restrict__ V,  const float* __restrict__ va,
    float* __restrict__ ctx)
{
  __shared__ float qa_s[16 * DKN];     // 4 KB
  __shared__ float sc[16 * LL];        // 24 KB score/prob tile
  __shared__ float va_s[DKN];
  __shared__ float rowmax[16];
  __shared__ float rowinv[16];

  const int tid = threadIdx.x;
  const int q0  = blockIdx.x * 16;
  const int h   = blockIdx.y;
  const int b   = blockIdx.z;
  const int base = b * LL * DD + h * DKN;  // + l*DD + d

  // Stage Qa tile and va into LDS via async global->LDS (bypasses VGPRs).
  for (int i = tid; i < 16 * DKN; i += 128) {
    const int q = i >> 6, d = i & 63;
    stage_b32_to_lds(Qa + base + (q0 + q) * DD + d, &qa_s[i]);
  }
  if (tid < DKN) stage_b32_to_lds(va + tid, &va_s[tid]);
  wait_async_lds();
  __syncthreads();

  // Scores: each thread owns k columns (3 per thread), 16 q-rows in registers.
  for (int k = tid; k < LL; k += 128) {
    const float* krow = Ka + base + k * DD;
    if (k + 128 < LL)                       // global_prefetch next Ka row
      __builtin_prefetch(krow + 128 * DD, 0, 1);
    float acc[16];
#pragma unroll
    for (int q = 0; q < 16; ++q) acc[q] = 0.f;
#pragma unroll 4
    for (int d4 = 0; d4 < DKN / 4; ++d4) {
      const v4f kv = *(const v4f*)(krow + d4 * 4);
#pragma unroll
      for (int j = 0; j < 4; ++j) {
        const int   d   = d4 * 4 + j;
        const float kd  = kv[j];
        const float vad = va_s[d];
#pragma unroll
        for (int q = 0; q < 16; ++q)          // qa_s read is a wave broadcast
          acc[q] += vad * fast_tanh(qa_s[q * DKN + d] + kd);
      }
    }
#pragma unroll
    for (int q = 0; q < 16; ++q) sc[q * LL + k] = acc[q];
  }
  __syncthreads();

  // Row softmax (16 rows; reductions are tiny vs. the tanh phase).
  if (tid < 16) {
    float m = -3.0e38f;
    for (int k = 0; k < LL; ++k) m = fmaxf(m, sc[tid * LL + k]);
    rowmax[tid] = m;
  }
  __syncthreads();
  for (int i = tid; i < 16 * LL; i += 128) {
    const int q = i / LL;
    sc[i] = __expf(sc[i] - rowmax[q]);
  }
  __syncthreads();
  if (tid < 16) {
    float s = 0.f;
    for (int k = 0; k < LL; ++k) s += sc[tid * LL + k];
    rowinv[tid] = 1.0f / s;
  }
  __syncthreads();

  // ctx[16 x 64] = P[16 x 384] @ V[384 x 64] via WMMA; 4 waves x 16-col tiles.
  const int lane  = tid & 31;
  const int wave  = tid >> 5;
  const int khalf = (lane >> 4) * 2;
  const int nn    = wave * 16 + (lane & 15);

  v8f acc = {};
  for (int kk = 0; kk < LL; kk += 4) {
    v2f a;
    a.x = sc[(lane & 15) * LL + kk + khalf];
    a.y = sc[(lane & 15) * LL + kk + khalf + 1];
    v2f bv;
    bv.x = V[base + (kk + khalf)     * DD + nn];
    bv.y = V[base + (kk + khalf + 1) * DD + nn];
    acc = WMMA_F32_4(a, bv, acc);
  }
#pragma unroll
  for (int r = 0; r < 8; ++r) {
    const int row = (lane >> 4) * 8 + r;
    ctx[base + (q0 + row) * DD + nn] = acc[r] * rowinv[row];
  }
}

// ---------------------------------------------------------------------------
extern "C" void kernel_launch(void* const* d_in, const int* in_sizes, int n_in,
                              void* d_out, int out_size, void* d_ws, size_t ws_size,
                              hipStream_t stream) {
  const float* query = (const float*)d_in[0];
  const float* key_  = (const float*)d_in[1];
  const float* value = (const float*)d_in[2];
  const float* Wq  = (const float*)d_in[3];
  const float* bq  = (const float*)d_in[4];
  const float* Wk  = (const float*)d_in[5];
  const float* bk  = (const float*)d_in[6];
  const float* Wv  = (const float*)d_in[7];
  const float* bv  = (const float*)d_in[8];
  const float* Wqa = (const float*)d_in[9];
  const float* bqa = (const float*)d_in[10];
  const float* Wka = (const float*)d_in[11];
  const float* bka = (const float*)d_in[12];
  const float* va  = (const float*)d_in[13];
  const float* Wo  = (const float*)d_in[14];
  const float* bo  = (const float*)d_in[15];

  float* ws = (float*)d_ws;
  const size_t NRD = (size_t)RR * DD;        // 393216 floats per buffer
  float* Qb  = ws + 0 * NRD;
  float* Kb  = ws + 1 * NRD;
  float* Vb  = ws + 2 * NRD;
  float* Qab = ws + 3 * NRD;
  float* Kab = ws + 4 * NRD;
  float* Ctx = ws + 5 * NRD;

  const dim3 blk(128);
  const dim3 gProj(RR / 16, DD / 64);        // 48 x 8
  const dim3 gHead(RR / 16, HH);             // 48 x 8
  const dim3 gAttn(LL / 16, HH, BB);         // 24 x 8 x 2

  gemm_bias_512<<<gProj, blk, 0, stream>>>(query, Wq, bq, Qb);
  gemm_bias_512<<<gProj, blk, 0, stream>>>(key_,  Wk, bk, Kb);
  gemm_bias_512<<<gProj, blk, 0, stream>>>(value, Wv, bv, Vb);
  head_gemm_64<<<gHead, blk, 0, stream>>>(Qb, Wqa, bqa, Qab);
  head_gemm_64<<<gHead, blk, 0, stream>>>(Kb, Wka, bka, Kab);
  additive_attn<<<gAttn, blk, 0, stream>>>(Qab, Kab, Vb, va, Ctx);
  gemm_bias_512<<<gProj, blk, 0, stream>>>(Ctx, Wo, bo, (float*)d_out);
}